// EquivariantBallUnpooling_1803886265812
// MI455X (gfx1250) — compile-verified
//
#include <hip/hip_runtime.h>
#include <math.h>

typedef __attribute__((ext_vector_type(2))) float v2f;
typedef __attribute__((ext_vector_type(4))) float v4f;
typedef __attribute__((ext_vector_type(8))) float v8f;

// ---------------- LDS layout constants ----------------
// MV kernel: cat_mv tile [128 parents][24 ch * 8 comps], padded row stride 196
//   (196*4B: 16B aligned; 196 mod 64 = 4 -> 4*m banks, conflict-free b128 reads)
#define MV_PARENTS   128
#define CAT_STRIDE   196
#define W_STRIDE     28     // 24 -> 28 (112B rows: 16B aligned, gcd(28,64)=4 -> conflict-free)
#define MV_CAT_FLOATS (MV_PARENTS * CAT_STRIDE)        // 25088
#define MV_W_FLOATS   (4 * 128 * W_STRIDE)             // 14336
#define MV_SMEM_BYTES ((MV_CAT_FLOATS + MV_W_FLOATS) * 4)   // 157,696 B

// S kernel: cat_s [128][40] and W_s [256][40], padded stride 44
#define S_STRIDE     44
#define S_CAT_FLOATS (128 * S_STRIDE)                  // 5632
#define S_W_FLOATS   (256 * S_STRIDE)                  // 11264
#define S_SMEM_BYTES ((S_CAT_FLOATS + S_W_FLOATS + 256) * 4) // 68,608 B

__device__ __forceinline__ v8f wmma_f32_4(v2f a, v2f b, v8f c) {
  // D = A(16x4 f32) * B(4x16 f32) + C(16x16 f32), full wave32 op
  return __builtin_amdgcn_wmma_f32_16x16x4_f32(false, a, false, b, (short)0, c,
                                               false, false);
}

// =====================================================================
// Multivector path:
//   y_mv[p,o,i] = sum_c cat_mv[p,c,i] * W_mv[grade(i)][o][c]   (o=0..127, c=0..23)
//   out_mv[p*1024 + o*8 + i] = child_x_mv[same] + y_mv
// =====================================================================
__global__ void __launch_bounds__(256)
mv_kernel(const float* __restrict__ x_mv, const float* __restrict__ pos,
          const float* __restrict__ child_x_mv, const float* __restrict__ child_pos,
          const float* __restrict__ W_mv, float* __restrict__ out_mv) {
  extern __shared__ float smem[];
  float* catT = smem;                    // [128][CAT_STRIDE] : [ch*8+comp] in 0..191
  float* Wm   = smem + MV_CAT_FLOATS;    // [(g*128+o)][W_STRIDE]

  const int tid   = threadIdx.x;
  const int pbase = blockIdx.x * MV_PARENTS;

  // ---- stage x_mv -> channels 0..15 (16384 floats = 4096 float4) ----
  {
    const v4f* src = (const v4f*)(x_mv + (size_t)pbase * 128);
    #pragma unroll
    for (int j = 0; j < 16; ++j) {
      int q  = tid + j * 256;
      int p  = q >> 5;            // 32 float4 per parent
      int w4 = q & 31;
      *(v4f*)(catT + p * CAT_STRIDE + w4 * 4) = src[q];
    }
  }
  // ---- stage translation-embedded rel -> channels 16..23 ----
  {
    #pragma unroll
    for (int j = 0; j < 4; ++j) {
      int c = tid + j * 256;      // 1024 children per WG
      int p = c >> 3, m = c & 7;
      size_t cg = ((size_t)(pbase + p) * 8 + m) * 3;
      size_t pg = (size_t)(pbase + p) * 3;
      float rx = child_pos[cg + 0] - pos[pg + 0];
      float ry = child_pos[cg + 1] - pos[pg + 1];
      float rz = child_pos[cg + 2] - pos[pg + 2];
      v4f lo = {0.f, rx, ry, rz};
      v4f hi = {0.f, 0.f, 0.f, 0.f};
      float* dst = catT + p * CAT_STRIDE + 128 + m * 8;
      *(v4f*)(dst)     = lo;
      *(v4f*)(dst + 4) = hi;
    }
  }
  // ---- stage W_mv: 512 rows (g,o) x 24 floats ----
  {
    #pragma unroll
    for (int j = 0; j < 2; ++j) {
      int row = tid + j * 256;
      const v4f* src = (const v4f*)(W_mv + (size_t)row * 24);  // 96B rows: aligned
      float* dst = Wm + row * W_STRIDE;
      #pragma unroll
      for (int q = 0; q < 6; ++q) *(v4f*)(dst + q * 4) = src[q];
    }
  }
  __syncthreads();

  const int lane = tid & 31;
  const int wv   = tid >> 5;        // 8 waves, 16 parents each
  const int ln   = lane & 15;
  const int half = lane >> 4;
  const int kb   = half * 2;        // A/B K-pair base per lane half (16x4 f32 layout)
  const float* arow = catT + (wv * 16 + ln) * CAT_STRIDE;

  for (int t = 0; t < 8; ++t) {     // 8 N-tiles of 16 over o=0..127
    v8f acc[8] = {};                // one 16x16 accumulator per component
    const int col = t * 16 + ln;
    #pragma unroll
    for (int kk = 0; kk < 6; ++kk) {
      const int k = kk * 4 + kb;    // channels k, k+1 for this lane half
      v4f A0lo = *(const v4f*)(arow + k * 8);
      v4f A0hi = *(const v4f*)(arow + k * 8 + 4);
      v4f A1lo = *(const v4f*)(arow + k * 8 + 8);
      v4f A1hi = *(const v4f*)(arow + k * 8 + 12);
      v2f b[4];
      #pragma unroll
      for (int g = 0; g < 4; ++g)
        b[g] = *(const v2f*)(Wm + (g * 128 + col) * W_STRIDE + k);
      #pragma unroll
      for (int i = 0; i < 8; ++i) { // grade-of-comp = [0,1,1,1,2,2,2,3]
        const int g = (i == 0) ? 0 : (i < 4) ? 1 : (i < 7) ? 2 : 3;
        v2f a;
        a.x = (i < 4) ? A0lo[i & 3] : A0hi[i & 3];
        a.y = (i < 4) ? A1lo[i & 3] : A1hi[i & 3];
        acc[i] = wmma_f32_4(a, b[g], acc[i]);   // 8-deep independent chains
      }
    }
    // epilogue: 8 comps of one (parent,o) are contiguous -> 2x b128 per lane
    #pragma unroll
    for (int r = 0; r < 8; ++r) {
      size_t addr = (size_t)(pbase + wv * 16 + half * 8 + r) * 1024 + (size_t)col * 8;
      v4f c0 = __builtin_nontemporal_load((const v4f*)(child_x_mv + addr));
      v4f c1 = __builtin_nontemporal_load((const v4f*)(child_x_mv + addr + 4));
      v4f y0, y1;
      #pragma unroll
      for (int j = 0; j < 4; ++j) {
        y0[j] = c0[j] + acc[j][r];
        y1[j] = c1[j] + acc[4 + j][r];
      }
      __builtin_nontemporal_store(y0, (v4f*)(out_mv + addr));
      __builtin_nontemporal_store(y1, (v4f*)(out_mv + addr + 4));
    }
  }
}

// =====================================================================
// Scalar path: y_s = cat_s @ W_s^T + b_s ;  out_s = child_x_s + y_s
//   cat_s[p] = [x_s[p] (32) || ||rel|| (8)]  ; out index = p*256 + col
// =====================================================================
__global__ void __launch_bounds__(256)
s_kernel(const float* __restrict__ x_s, const float* __restrict__ pos,
         const float* __restrict__ child_x_s, const float* __restrict__ child_pos,
         const float* __restrict__ W_s, const float* __restrict__ b_s,
         float* __restrict__ out_s) {
  extern __shared__ float smem[];
  float* catS = smem;                     // [128][S_STRIDE]
  float* Ws   = smem + S_CAT_FLOATS;      // [256][S_STRIDE]
  float* bs   = Ws + S_W_FLOATS;          // [256]

  const int tid   = threadIdx.x;
  const int pbase = blockIdx.x * 128;

  // ---- stage x_s (128*32 floats = 1024 float4) ----
  {
    const v4f* src = (const v4f*)(x_s + (size_t)pbase * 32);
    #pragma unroll
    for (int j = 0; j < 4; ++j) {
      int q = tid + j * 256;
      int p = q >> 3, w4 = q & 7;
      *(v4f*)(catS + p * S_STRIDE + w4 * 4) = src[q];
    }
  }
  // ---- stage ||rel|| -> cols 32..39 ----
  {
    #pragma unroll
    for (int j = 0; j < 4; ++j) {
      int c = tid + j * 256;
      int p = c >> 3, m = c & 7;
      size_t cg = ((size_t)(pbase + p) * 8 + m) * 3;
      size_t pg = (size_t)(pbase + p) * 3;
      float rx = child_pos[cg + 0] - pos[pg + 0];
      float ry = child_pos[cg + 1] - pos[pg + 1];
      float rz = child_pos[cg + 2] - pos[pg + 2];
      catS[p * S_STRIDE + 32 + m] = sqrtf(rx * rx + ry * ry + rz * rz);
    }
  }
  // ---- stage W_s (256 rows x 40, 160B rows: float4-aligned) + bias ----
  {
    const v4f* src = (const v4f*)(W_s + (size_t)tid * 40);
    float* dst = Ws + tid * S_STRIDE;
    #pragma unroll
    for (int q = 0; q < 10; ++q) *(v4f*)(dst + q * 4) = src[q];
    bs[tid] = b_s[tid];
  }
  __syncthreads();

  const int lane = tid & 31;
  const int wv   = tid >> 5;
  const int ln   = lane & 15;
  const int half = lane >> 4;
  const int kb   = half * 2;
  const float* arow = catS + (wv * 16 + ln) * S_STRIDE;

  for (int t = 0; t < 16; t += 4) {   // 16 N-tiles over 256 cols, 4 at a time
    v8f acc0 = {}, acc1 = {}, acc2 = {}, acc3 = {};
    const int c0 = t * 16 + ln, c1 = c0 + 16, c2 = c0 + 32, c3 = c0 + 48;
    #pragma unroll
    for (int kk = 0; kk < 10; ++kk) {  // K = 40
      const int k = kk * 4 + kb;
      v2f a  = *(const v2f*)(arow + k);
      v2f b0 = *(const v2f*)(Ws + c0 * S_STRIDE + k);
      v2f b1 = *(const v2f*)(Ws + c1 * S_STRIDE + k);
      v2f b2 = *(const v2f*)(Ws + c2 * S_STRIDE + k);
      v2f b3 = *(const v2f*)(Ws + c3 * S_STRIDE + k);
      acc0 = wmma_f32_4(a, b0, acc0);
      acc1 = wmma_f32_4(a, b1, acc1);
      acc2 = wmma_f32_4(a, b2, acc2);
      acc3 = wmma_f32_4(a, b3, acc3);
    }
    const float bias0 = bs[c0], bias1 = bs[c1], bias2 = bs[c2], bias3 = bs[c3];
    #pragma unroll
    for (int r = 0; r < 8; ++r) {
      size_t base = (size_t)(pbase + wv * 16 + half * 8 + r) * 256;
      float d0 = __builtin_nontemporal_load(child_x_s + base + c0) + acc0[r] + bias0;
      float d1 = __builtin_nontemporal_load(child_x_s + base + c1) + acc1[r] + bias1;
      float d2 = __builtin_nontemporal_load(child_x_s + base + c2) + acc2[r] + bias2;
      float d3 = __builtin_nontemporal_load(child_x_s + base + c3) + acc3[r] + bias3;
      __builtin_nontemporal_store(d0, out_s + base + c0);
      __builtin_nontemporal_store(d1, out_s + base + c1);
      __builtin_nontemporal_store(d2, out_s + base + c2);
      __builtin_nontemporal_store(d3, out_s + base + c3);
    }
  }
}

extern "C" void kernel_launch(void* const* d_in, const int* in_sizes, int n_in,
                              void* d_out, int out_size, void* d_ws, size_t ws_size,
                              hipStream_t stream) {
  const float* x_mv       = (const float*)d_in[0];
  const float* x_s        = (const float*)d_in[1];
  const float* pos        = (const float*)d_in[2];
  const float* child_x_mv = (const float*)d_in[3];
  const float* child_x_s  = (const float*)d_in[4];
  const float* child_pos  = (const float*)d_in[5];
  const float* W_mv       = (const float*)d_in[6];
  const float* W_s        = (const float*)d_in[7];
  const float* b_s        = (const float*)d_in[8];
  (void)d_ws; (void)ws_size; (void)n_in;

  const int n = in_sizes[0] / 128;          // x_mv is (n,16,8)
  float* out_mv = (float*)d_out;            // (n*8,16,8) = n*1024 floats
  float* out_s  = (float*)d_out + (size_t)n * 1024;  // (n*8,32) = n*256 floats
  const int blocks = n / 128;               // n = 65536 -> 512 WGs

  mv_kernel<<<blocks, 256, MV_SMEM_BYTES, stream>>>(x_mv, pos, child_x_mv,
                                                    child_pos, W_mv, out_mv);
  s_kernel<<<blocks, 256, S_SMEM_BYTES, stream>>>(x_s, pos, child_x_s,
                                                  child_pos, W_s, b_s, out_s);
  (void)out_size;
}